// HyperGCN_2594160246963
// MI455X (gfx1250) — compile-verified
//
#include <hip/hip_runtime.h>

typedef float v2f __attribute__((ext_vector_type(2)));
typedef float v8f __attribute__((ext_vector_type(8)));

// ---------------------------------------------------------------------------
// Pre-swizzle weight matrix W[K x nvalid] (row-major) into per-lane WMMA B
// operand order, zero-padded to ntn*16 columns:
//   Bsw[((tn*ksteps + s)*32 + lane)*2 + j] = W[4s + (lane>>4)*2 + j, tn*16 + (lane&15)]
// so the GEMM inner loop does one coalesced b64 load per B operand.
// ---------------------------------------------------------------------------
__global__ void swizzleB(const float* __restrict__ W, float* __restrict__ Bsw,
                         int K, int nvalid, int ntn)
{
    int i = blockIdx.x * blockDim.x + threadIdx.x;
    int ksteps = K >> 2;
    int total  = ntn * ksteps * 64;
    if (i >= total) return;
    int j    = i & 1;
    int lane = (i >> 1) & 31;
    int s    = (i >> 6) % ksteps;
    int tn   = i / (ksteps * 64);
    int kk   = 4 * s + ((lane >> 4) << 1) + j;
    int c    = tn * 16 + (lane & 15);
    Bsw[i] = (c < nvalid) ? W[(size_t)kk * nvalid + c] : 0.0f;
}

// ---------------------------------------------------------------------------
// fp32 WMMA GEMM: C[M x NTN*16] = op(A)[M x K] @ Bsw   (op = optional ReLU)
// One wave32 owns one 16-row tile and ALL NTN column tiles, so the A operand
// (the HBM-bound stream) is loaded exactly once. K multiple of 4, M mult of 16.
// ---------------------------------------------------------------------------
template<int NTN, bool RELU>
__global__ void gemm_wmma(const float* __restrict__ A, const float* __restrict__ Bsw,
                          float* __restrict__ C, int M, int K)
{
    int tid  = blockIdx.x * blockDim.x + threadIdx.x;
    int wid  = tid >> 5;
    int lane = threadIdx.x & 31;
    int ntm  = M >> 4;
    if (wid >= ntm) return;               // wave-uniform exit, EXEC all-ones at WMMA

    int lane16 = lane & 15;
    int khalf  = (lane >> 4) << 1;        // 0 or 2: this half-wave's K pair
    int row    = (wid << 4) + lane16;

    const float* arow = A + (size_t)row * K + khalf;
    __builtin_prefetch(arow, 0, 3);       // global_prefetch_b8

    const int ksteps = K >> 2;
    const float* bptr = Bsw + (size_t)lane * 2;

    v8f acc[NTN];
    #pragma unroll
    for (int t = 0; t < NTN; ++t) acc[t] = {};

    #pragma unroll 4
    for (int s = 0; s < ksteps; ++s) {
        v2f a = *(const v2f*)(arow + 4 * s);          // global_load_b64
        if (RELU) { a.x = fmaxf(a.x, 0.0f); a.y = fmaxf(a.y, 0.0f); }
        #pragma unroll
        for (int t = 0; t < NTN; ++t) {
            v2f b = *(const v2f*)(bptr + (size_t)(t * ksteps + s) * 64);  // b64, coalesced
            acc[t] = __builtin_amdgcn_wmma_f32_16x16x4_f32(
                         false, a, false, b, (short)0, acc[t], false, false);
        }
    }

    const int ldC  = NTN * 16;
    int rbase = (wid << 4) + ((lane >> 4) << 3);      // VGPR v -> row v (+8 upper half)
    #pragma unroll
    for (int t = 0; t < NTN; ++t) {
        float* cp = C + (size_t)rbase * ldC + t * 16 + lane16;
        #pragma unroll
        for (int v = 0; v < 8; ++v)
            cp[(size_t)v * ldC] = acc[t][v];
    }
}

// ---------------------------------------------------------------------------
// dst[i] = bias[i % F]
// ---------------------------------------------------------------------------
__global__ void fill_bias(float* __restrict__ dst, const float* __restrict__ bias,
                          int total, int F)
{
    int i = blockIdx.x * blockDim.x + threadIdx.x;
    if (i < total) dst[i] = bias[i % F];
}

// ---------------------------------------------------------------------------
// COO scatter: dst[row[e], f] += w[e] * src[col[e], f]
// src has row stride sld (lets layer 2 read the padded N x 48 product),
// dst has row stride F. Hardware fp32 atomics; src/dst live in L2.
// ---------------------------------------------------------------------------
template<int F>
__global__ void scatter_edges(const int* __restrict__ row, const int* __restrict__ col,
                              const float* __restrict__ w, const float* __restrict__ src,
                              float* __restrict__ dst, int nnz, int sld)
{
    int tid  = blockIdx.x * blockDim.x + threadIdx.x;
    int wid  = tid >> 5;
    int lane = threadIdx.x & 31;

    if constexpr (F == 16) {
        int e = (wid << 1) + (lane >> 4);             // two edges per wave
        if (e >= nnz) return;
        int f = lane & 15;
        int c = col[e], r = row[e];
        float v = w[e] * src[(size_t)c * sld + f];
        unsafeAtomicAdd(dst + (size_t)r * 16 + f, v);
    } else if constexpr (F == 32) {
        int e = wid;                                   // one edge per wave, lane = feature
        if (e >= nnz) return;
        int c = col[e], r = row[e];
        float wv = w[e];
        unsafeAtomicAdd(dst + (size_t)r * 32 + lane, wv * src[(size_t)c * sld + lane]);
    } else {                                           // F == 40
        int e = wid;
        if (e >= nnz) return;
        int c = col[e], r = row[e];
        float wv = w[e];
        unsafeAtomicAdd(dst + (size_t)r * 40 + lane, wv * src[(size_t)c * sld + lane]);
        if (lane < 8) {
            int f = lane + 32;
            unsafeAtomicAdd(dst + (size_t)r * 40 + f, wv * src[(size_t)c * sld + f]);
        }
    }
}

__global__ void relu_inplace(float* __restrict__ p, int total)
{
    int i = blockIdx.x * blockDim.x + threadIdx.x;
    if (i < total) p[i] = fmaxf(p[i], 0.0f);
}

// ---------------------------------------------------------------------------
// Host-side orchestration
// ---------------------------------------------------------------------------
extern "C" void kernel_launch(void* const* d_in, const int* in_sizes, int n_in,
                              void* d_out, int out_size, void* d_ws, size_t ws_size,
                              hipStream_t stream)
{
    const float* x    = (const float*)d_in[0];
    const int*   erow = (const int*)  d_in[1];
    const int*   ecol = (const int*)  d_in[2];
    const float* ew   = (const float*)d_in[3];
    const float* W0   = (const float*)d_in[4];
    const float* b0   = (const float*)d_in[5];
    const float* W1   = (const float*)d_in[6];
    const float* b1   = (const float*)d_in[7];
    const float* W2   = (const float*)d_in[8];
    const float* b2   = (const float*)d_in[9];

    const int D = 512, H1 = 32, H2 = 16, Cout = 40;
    const int N   = in_sizes[0] / D;      // 100000
    const int NNZ = in_sizes[1];          // 3200000

    // --- workspace layout (floats) ---
    float* ws   = (float*)d_ws;
    const size_t szB0 = (size_t)2 * (D  / 4) * 64;   // 16384
    const size_t szB1 = (size_t)1 * (H1 / 4) * 64;   // 512
    const size_t szB2 = (size_t)3 * (H2 / 4) * 64;   // 768
    float* Bsw0 = ws;
    float* Bsw1 = Bsw0 + szB0;
    float* Bsw2 = Bsw1 + szB1;
    float* base = Bsw2 + szB2;
    float* T0   = base;                    // N*32
    float* A0   = base + (size_t)N * 32;   // N*32
    float* T1   = base;                    // N*16 (reuses T0 slot)
    float* A1   = base + (size_t)N * 32;   // N*16 (reuses A0 slot)
    float* T2   = base + (size_t)N * 48;   // N*48 (padded: 3 col tiles)
    float* out  = (float*)d_out;           // N*40

    const int BLK = 256;
    auto blocks = [&](long long threads) { return (int)((threads + BLK - 1) / BLK); };

    // --- weight pre-swizzle (tiny) ---
    swizzleB<<<blocks((long long)szB0), BLK, 0, stream>>>(W0, Bsw0, D,  H1,   2);
    swizzleB<<<blocks((long long)szB1), BLK, 0, stream>>>(W1, Bsw1, H1, H2,   1);
    swizzleB<<<blocks((long long)szB2), BLK, 0, stream>>>(W2, Bsw2, H2, Cout, 3);

    const long long gthr = (long long)(N / 16) * 32;   // one wave per 16-row tile

    // ---- Layer 0: T0 = X @ W0 ----
    gemm_wmma<2, false><<<blocks(gthr), BLK, 0, stream>>>(x, Bsw0, T0, N, D);
    fill_bias<<<blocks((long long)N * H1), BLK, 0, stream>>>(A0, b0, N * H1, H1);
    scatter_edges<32><<<blocks((long long)NNZ * 32), BLK, 0, stream>>>(erow, ecol, ew, T0, A0, NNZ, 32);

    // ---- Layer 1: T1 = relu(A0) @ W1 ----
    gemm_wmma<1, true><<<blocks(gthr), BLK, 0, stream>>>(A0, Bsw1, T1, N, H1);
    fill_bias<<<blocks((long long)N * H2), BLK, 0, stream>>>(A1, b1, N * H2, H2);
    {
        long long waves = (NNZ + 1) / 2;
        scatter_edges<16><<<blocks(waves * 32), BLK, 0, stream>>>(erow, ecol, ew, T1, A1, NNZ, 16);
    }

    // ---- Layer 2: T2 = relu(A1) @ W2 (padded to 48 cols) ----
    gemm_wmma<3, true><<<blocks(gthr), BLK, 0, stream>>>(A1, Bsw2, T2, N, H2);
    fill_bias<<<blocks((long long)N * Cout), BLK, 0, stream>>>(out, b2, N * Cout, Cout);
    scatter_edges<40><<<blocks((long long)NNZ * 32), BLK, 0, stream>>>(erow, ecol, ew, T2, out, NNZ, 48);
    relu_inplace<<<blocks((long long)N * Cout), BLK, 0, stream>>>(out, N * Cout);
}